// SimpleGNN_35296041238623
// MI455X (gfx1250) — compile-verified
//
#include <hip/hip_runtime.h>
#include <hip/hip_bf16.h>

typedef __attribute__((ext_vector_type(2))) float v2f;
typedef __attribute__((ext_vector_type(8))) float v8f;

#define NEG_SLOPE 0.2f
#define EPS_DEN 1e-16f
#define FDIM 64

// Order-preserving float<->uint map so atomicMax(u32) implements float segment-max.
__device__ __forceinline__ unsigned fenc(float f) {
  unsigned b = __float_as_uint(f);
  return (b & 0x80000000u) ? ~b : (b | 0x80000000u);
}
__device__ __forceinline__ float fdec(unsigned k) {
  unsigned b = (k & 0x80000000u) ? (k & 0x7fffffffu) : ~k;
  return __uint_as_float(b);
}

// xl = X @ Wl, xr = X @ Wr  (X: [N,K] row-major, W: [K,64] row-major)
// One 16-row tile per block; 4 waves each own a 16-col tile; A fragment shared
// between the two accumulators. Exact fp32 via V_WMMA_F32_16X16X4_F32.
__global__ __launch_bounds__(128) void gatv2_gemm_dual(
    const float* __restrict__ X, const float* __restrict__ Wl,
    const float* __restrict__ Wr, float* __restrict__ xl,
    float* __restrict__ xr, int N, int K) {
  const int lane   = threadIdx.x & 31;
  const int wave   = threadIdx.x >> 5;
  const int laneHi = lane >> 4;      // 0: K pair {k,k+1}, 1: {k+2,k+3}
  const int lane16 = lane & 15;
  const int rowTile = blockIdx.x;
  const int colBase = wave * 16;

  int row  = rowTile * 16 + lane16;
  int rowc = row < N ? row : N - 1;  // clamp loads, keep EXEC uniform for WMMA

  v8f accL = {};
  v8f accR = {};
  const float* xrow = X + (size_t)rowc * K;

  for (int k = 0; k < K; k += 4) {
    const int kk = k + 2 * laneHi;
    // A fragment (16x4 f32): lane<16 holds row M=lane16, K={k,k+1};
    // lanes 16..31 hold K={k+2,k+3}. Contiguous -> one 8B load.
    v2f a = *(const v2f*)(xrow + kk);
    // B fragment (4x16 f32): lane = column N, VGPR pair = K within lane half.
    v2f bl, br;
    bl[0] = Wl[(size_t)kk * FDIM + colBase + lane16];
    bl[1] = Wl[(size_t)(kk + 1) * FDIM + colBase + lane16];
    br[0] = Wr[(size_t)kk * FDIM + colBase + lane16];
    br[1] = Wr[(size_t)(kk + 1) * FDIM + colBase + lane16];
    accL = __builtin_amdgcn_wmma_f32_16x16x4_f32(false, a, false, bl,
                                                 (short)0, accL, false, false);
    accR = __builtin_amdgcn_wmma_f32_16x16x4_f32(false, a, false, br,
                                                 (short)0, accR, false, false);
  }

  // C/D layout: VGPR v -> M = v + 8*laneHi, lane16 -> N column.
#pragma unroll
  for (int v = 0; v < 8; ++v) {
    int m = v + 8 * laneHi;
    int r = rowTile * 16 + m;
    if (r < N) {
      xl[(size_t)r * FDIM + colBase + lane16] = accL[v];
      xr[(size_t)r * FDIM + colBase + lane16] = accR[v];
    }
  }
}

// One wave per edge: e = att . leaky_relu(xl[src] + xr[dst]); atomic segment max.
__global__ __launch_bounds__(256) void gatv2_edge_logit(
    const float* __restrict__ xl, const float* __restrict__ xr,
    const int* __restrict__ esrc, const int* __restrict__ edst,
    const float* __restrict__ att, float* __restrict__ e,
    unsigned* __restrict__ mkey, int E, int ET) {
  int eid = blockIdx.x * 8 + (threadIdx.x >> 5);
  if (eid >= ET) return;
  int lane = threadIdx.x & 31;
  int s, d;
  if (eid < E) { s = esrc[eid]; d = edst[eid]; }
  else         { s = d = eid - E; }  // self loop

  v2f a  = *(const v2f*)(xl + (size_t)s * FDIM + 2 * lane);
  v2f b  = *(const v2f*)(xr + (size_t)d * FDIM + 2 * lane);
  v2f at = *(const v2f*)(att + 2 * lane);
  float t0 = a[0] + b[0]; t0 = t0 > 0.f ? t0 : NEG_SLOPE * t0;
  float t1 = a[1] + b[1]; t1 = t1 > 0.f ? t1 : NEG_SLOPE * t1;
  float sum = t0 * at[0] + t1 * at[1];
#pragma unroll
  for (int off = 16; off; off >>= 1) sum += __shfl_xor(sum, off, 32);
  if (lane == 0) {
    e[eid] = sum;
    atomicMax(mkey + d, fenc(sum));
  }
}

// One thread per edge: ex = exp(e - max[dst]); denom[dst] += ex.
__global__ __launch_bounds__(256) void gatv2_edge_exp(
    const float* __restrict__ e, const int* __restrict__ edst,
    const unsigned* __restrict__ mkey, float* __restrict__ ex,
    float* __restrict__ denom, int E, int ET) {
  int eid = blockIdx.x * 256 + threadIdx.x;
  if (eid >= ET) return;
  int d = (eid < E) ? edst[eid] : (eid - E);
  float v = __expf(e[eid] - fdec(mkey[d]));
  ex[eid] = v;
  atomicAdd(denom + d, v);
}

// One wave per edge: agg[dst] += alpha * xl[src] (global_atomic_add_f32).
__global__ __launch_bounds__(256) void gatv2_edge_aggregate(
    const float* __restrict__ xl, const float* __restrict__ ex,
    const float* __restrict__ denom, const int* __restrict__ esrc,
    const int* __restrict__ edst, float* __restrict__ agg, int E, int ET) {
  int eid = blockIdx.x * 8 + (threadIdx.x >> 5);
  if (eid >= ET) return;
  int lane = threadIdx.x & 31;
  int s, d;
  if (eid < E) { s = esrc[eid]; d = edst[eid]; }
  else         { s = d = eid - E; }

  float alpha = ex[eid] / (denom[d] + EPS_DEN);
  v2f a = *(const v2f*)(xl + (size_t)s * FDIM + 2 * lane);
  float* dp = agg + (size_t)d * FDIM + 2 * lane;
  atomicAdd(dp, alpha * a[0]);
  atomicAdd(dp + 1, alpha * a[1]);
}

// out = (relu ? max(0, agg + bias) : agg + bias)
__global__ __launch_bounds__(256) void gatv2_finalize(
    const float* __restrict__ agg, const float* __restrict__ bias,
    float* __restrict__ out, size_t n, int relu) {
  size_t i = (size_t)blockIdx.x * 256 + threadIdx.x;
  if (i >= n) return;
  float v = agg[i] + bias[i & (FDIM - 1)];
  out[i] = (relu && v < 0.f) ? 0.f : v;
}

extern "C" void kernel_launch(void* const* d_in, const int* in_sizes, int n_in,
                              void* d_out, int out_size, void* d_ws, size_t ws_size,
                              hipStream_t stream) {
  const float* x    = (const float*)d_in[0];
  const int*   ei   = (const int*)d_in[1];   // [2,E]: row0=src, row1=dst
  const float* W1l  = (const float*)d_in[2];
  const float* W1r  = (const float*)d_in[3];
  const float* att1 = (const float*)d_in[4];
  const float* b1   = (const float*)d_in[5];
  const float* W2l  = (const float*)d_in[6];
  const float* W2r  = (const float*)d_in[7];
  const float* att2 = (const float*)d_in[8];
  const float* b2   = (const float*)d_in[9];

  const int N  = in_sizes[0] / 128;   // 50000
  const int E  = in_sizes[1] / 2;     // 800000
  const int ET = E + N;               // + self loops

  // Workspace layout (floats): xl | xr | h | agg | e | ex | mkey | denom
  float* ws = (float*)d_ws;
  const size_t NF = (size_t)N * FDIM;
  float*    xl    = ws;
  float*    xr    = xl + NF;
  float*    h     = xr + NF;
  float*    agg   = h + NF;
  float*    e     = agg + NF;
  float*    ex    = e + ET;
  unsigned* mkey  = (unsigned*)(ex + ET);
  float*    denom = (float*)(mkey + N);

  const int rowTiles   = (N + 15) / 16;
  const int edgeWaveBl = (ET + 7) / 8;
  const int edgeThrBl  = (ET + 255) / 256;
  const int nfBl       = (int)((NF + 255) / 256);
  const int* esrc = ei;
  const int* edst = ei + E;

  // ---- Layer 1 (K = 128) ----
  hipMemsetAsync(mkey, 0, sizeof(unsigned) * N, stream);
  hipMemsetAsync(denom, 0, sizeof(float) * N, stream);
  hipMemsetAsync(agg, 0, sizeof(float) * NF, stream);
  gatv2_gemm_dual<<<rowTiles, 128, 0, stream>>>(x, W1l, W1r, xl, xr, N, 128);
  gatv2_edge_logit<<<edgeWaveBl, 256, 0, stream>>>(xl, xr, esrc, edst, att1, e, mkey, E, ET);
  gatv2_edge_exp<<<edgeThrBl, 256, 0, stream>>>(e, edst, mkey, ex, denom, E, ET);
  gatv2_edge_aggregate<<<edgeWaveBl, 256, 0, stream>>>(xl, ex, denom, esrc, edst, agg, E, ET);
  gatv2_finalize<<<nfBl, 256, 0, stream>>>(agg, b1, h, NF, 1);

  // ---- Layer 2 (K = 64) ----
  hipMemsetAsync(mkey, 0, sizeof(unsigned) * N, stream);
  hipMemsetAsync(denom, 0, sizeof(float) * N, stream);
  hipMemsetAsync(agg, 0, sizeof(float) * NF, stream);
  gatv2_gemm_dual<<<rowTiles, 128, 0, stream>>>(h, W2l, W2r, xl, xr, N, 64);
  gatv2_edge_logit<<<edgeWaveBl, 256, 0, stream>>>(xl, xr, esrc, edst, att2, e, mkey, E, ET);
  gatv2_edge_exp<<<edgeThrBl, 256, 0, stream>>>(e, edst, mkey, ex, denom, E, ET);
  gatv2_edge_aggregate<<<edgeWaveBl, 256, 0, stream>>>(xl, ex, denom, esrc, edst, agg, E, ET);
  gatv2_finalize<<<nfBl, 256, 0, stream>>>(agg, b2, (float*)d_out, NF, 0);
}